// MessageGraphConvolution_76914274337236
// MI455X (gfx1250) — compile-verified
//
#include <hip/hip_runtime.h>

typedef __attribute__((ext_vector_type(2))) float v2f;
typedef __attribute__((ext_vector_type(8))) float v8f;

#define IN_FEAT 128
#define OUT_FEAT 128

// ---------------------------------------------------------------------------
// Kernel 1: zero the workspace (agg + deg are contiguous in d_ws), b128 stores
// ---------------------------------------------------------------------------
__global__ void mgc_zero_kernel(float* __restrict__ p, int n) {
    int n4 = n >> 2;
    float4* p4 = (float4*)p;
    int i = blockIdx.x * blockDim.x + threadIdx.x;
    int stride = gridDim.x * blockDim.x;
    const float4 z = {0.f, 0.f, 0.f, 0.f};
    for (int j = i; j < n4; j += stride) p4[j] = z;
    // tail (n not multiple of 4)
    for (int j = (n4 << 2) + i; j < n; j += stride) p[j] = 0.0f;
}

// ---------------------------------------------------------------------------
// Kernel 2: edge-parallel scatter-add. One wave32 per edge:
// 32 lanes x float4 = one full 512B source row, coalesced; 4 f32 atomics/lane
// into agg[dst]. x and agg both live in the 192MB L2, so these atomics are
// L2-resident. Edge id is forced wave-uniform so src/dst become scalar loads.
// ---------------------------------------------------------------------------
__global__ void mgc_scatter_kernel(const float* __restrict__ x,
                                   const int* __restrict__ src,
                                   const int* __restrict__ dst,
                                   float* __restrict__ agg,
                                   float* __restrict__ deg,
                                   int num_edges) {
    // 8 waves per 256-thread block; e is uniform across the wave.
    int e = __builtin_amdgcn_readfirstlane(
                (int)(blockIdx.x * 8 + (threadIdx.x >> 5)));
    int lane = threadIdx.x & 31;
    if (e >= num_edges) return;   // scalar branch

    int s = src[e];               // scalar (SMEM) loads
    int d = dst[e];

    const float4* xrow = (const float4*)(x + (size_t)s * IN_FEAT);
    float4 v = xrow[lane];        // 512B coalesced per wave

    float* arow = agg + (size_t)d * IN_FEAT + lane * 4;
    atomicAdd(arow + 0, v.x);
    atomicAdd(arow + 1, v.y);
    atomicAdd(arow + 2, v.z);
    atomicAdd(arow + 3, v.w);

    if (lane == 0) atomicAdd(deg + d, 1.0f);
}

// ---------------------------------------------------------------------------
// Kernel 2.5: deg -> 1/deg in place (zero-degree -> 1). One precise IEEE
// division per node (10K total) so the WMMA kernel never divides.
// ---------------------------------------------------------------------------
__global__ void mgc_invdeg_kernel(float* __restrict__ deg, int num_nodes) {
    int i = blockIdx.x * blockDim.x + threadIdx.x;
    int stride = gridDim.x * blockDim.x;
    for (; i < num_nodes; i += stride) {
        float d = deg[i];
        deg[i] = d > 0.0f ? 1.0f / d : 1.0f;
    }
}

// ---------------------------------------------------------------------------
// Kernel 3: out = (agg * invd) @ W^T + x @ B^T using V_WMMA_F32_16X16X4_F32.
// One wave per 16x16 output tile. Row-normalization commutes with the GEMM,
// so chain 1 runs unnormalized (pure load->wmma inner loop), then the 8
// accumulator VGPRs are scaled once per row by the precomputed 1/deg, then
// chain 2 accumulates on top. Operand layouts per ISA 7.12.2:
//   A 16x4 f32 : lane (m = lane&15) holds K = 2*half, 2*half+1
//   B 4x16 f32 : lane (n = lane&15) holds B[k][n] = W[n][k]  (contiguous in W)
//   C/D 16x16  : VGPR r -> rows r (lanes 0-15) and r+8 (lanes 16-31)
// All 32 lanes stay active through the WMMA chains (EXEC must be all 1s).
// ---------------------------------------------------------------------------
__global__ void mgc_gemm_wmma_kernel(const float* __restrict__ agg,
                                     const float* __restrict__ invdeg,
                                     const float* __restrict__ x,
                                     const float* __restrict__ W,
                                     const float* __restrict__ Bm,
                                     float* __restrict__ out,
                                     int num_nodes) {
    int wave = threadIdx.x >> 5;      // 8 waves -> 8 column tiles (128 cols)
    int lane = threadIdx.x & 31;
    int half = lane >> 4;             // 0: K 0/1 side, 1: K 2/3 side
    int lm   = lane & 15;

    int row0 = blockIdx.x * 16;
    int col0 = wave * 16;

    // Clamp rows for loads so EXEC stays all-ones in a ragged last tile.
    int arow = row0 + lm;
    int lrow = arow < num_nodes ? arow : num_nodes - 1;
    int brow = col0 + lm;             // row of W / B (output feature)

    // Per-accumulator-register inverse degree (precomputed): c VGPR r holds
    // output rows (row0 + r + half*8). Load up-front to overlap latency.
    float invd[8];
#pragma unroll
    for (int r = 0; r < 8; ++r) {
        int rr = row0 + r + half * 8;
        rr = rr < num_nodes ? rr : num_nodes - 1;
        invd[r] = invdeg[rr];
    }

    v8f c = {0.f, 0.f, 0.f, 0.f, 0.f, 0.f, 0.f, 0.f};

    // ---- chain 1: agg @ W^T (unnormalized; pure load->wmma loop) ----
    {
        const float* ap = agg + (size_t)lrow * IN_FEAT + 2 * half;
        const float* wp = W   + (size_t)brow * IN_FEAT + 2 * half;
#pragma unroll 8
        for (int k = 0; k < IN_FEAT; k += 4) {
            v2f a = { ap[k], ap[k + 1] };
            v2f b = { wp[k], wp[k + 1] };
            c = __builtin_amdgcn_wmma_f32_16x16x4_f32(
                    false, a, false, b, (short)0, c, false, false);
        }
    }

    // ---- scale accumulator rows by precomputed 1/deg (8 v_mul_f32) ----
#pragma unroll
    for (int r = 0; r < 8; ++r) c[r] *= invd[r];

    // ---- chain 2: x @ B^T ----
    {
        const float* xp = x  + (size_t)lrow * IN_FEAT + 2 * half;
        const float* bp = Bm + (size_t)brow * IN_FEAT + 2 * half;
#pragma unroll 8
        for (int k = 0; k < IN_FEAT; k += 4) {
            v2f a = { xp[k], xp[k + 1] };
            v2f b = { bp[k], bp[k + 1] };
            c = __builtin_amdgcn_wmma_f32_16x16x4_f32(
                    false, a, false, b, (short)0, c, false, false);
        }
    }

    // ---- store D: VGPR r -> row (row0 + r + half*8), col (col0 + lm) ----
    int ocol = col0 + lm;
#pragma unroll
    for (int r = 0; r < 8; ++r) {
        int orow = row0 + r + half * 8;
        if (orow < num_nodes)
            out[(size_t)orow * OUT_FEAT + ocol] = c[r];
    }
}

// ---------------------------------------------------------------------------
extern "C" void kernel_launch(void* const* d_in, const int* in_sizes, int n_in,
                              void* d_out, int out_size, void* d_ws, size_t ws_size,
                              hipStream_t stream) {
    const float* x  = (const float*)d_in[0];
    const int*   ei = (const int*)d_in[1];   // [2, num_edges] flat
    const float* W  = (const float*)d_in[2];
    const float* Bm = (const float*)d_in[3];
    float*       out = (float*)d_out;

    int num_nodes = in_sizes[0] / IN_FEAT;
    int num_edges = in_sizes[1] / 2;

    const int* src = ei;
    const int* dst = ei + num_edges;

    float* agg = (float*)d_ws;                       // num_nodes * 128 f32
    float* deg = agg + (size_t)num_nodes * IN_FEAT;  // num_nodes f32

    // 1) zero agg + deg (contiguous)
    int ztotal = num_nodes * (IN_FEAT + 1);
    mgc_zero_kernel<<<640, 256, 0, stream>>>(agg, ztotal);

    // 2) scatter-add: one wave per edge (8 waves per block)
    int eblocks = (num_edges + 7) / 8;
    mgc_scatter_kernel<<<eblocks, 256, 0, stream>>>(x, src, dst, agg, deg, num_edges);

    // 2.5) deg -> 1/deg (in place)
    int dblocks = (num_nodes + 255) / 256;
    mgc_invdeg_kernel<<<dblocks, 256, 0, stream>>>(deg, num_nodes);

    // 3) normalize + dual GEMM via f32 WMMA: one block per 16-row tile
    int rtiles = (num_nodes + 15) / 16;
    mgc_gemm_wmma_kernel<<<rtiles, 256, 0, stream>>>(agg, deg, x, W, Bm, out, num_nodes);
}